// MultiHeadAttention_41351945126163
// MI455X (gfx1250) — compile-verified
//
#include <hip/hip_runtime.h>
#include <hip/hip_bf16.h>

// MHA for MI455X (gfx1250, wave32). fp32 WMMA (V_WMMA_F32_16X16X4_F32) for all
// matmuls (matches the fp32 reference; the problem is memory/issue bound so
// the lower fp32 matrix rate costs nothing). Flash-attention streaming with
// GLOBAL_LOAD_ASYNC_TO_LDS_B128 double buffering (ASYNCcnt), bank-conflict-
// free padded LDS tiles, b64 operand loads via pre-transposed W and V, and
// launch_bounds(256,1) so the register allocator can keep all 12 B-fragment
// loads in flight (12-load clauses + staggered waits instead of MLP=1).
//
// Shapes: B=4 S=2048 D=512 H=8 DH=64.

typedef float v2f __attribute__((ext_vector_type(2)));
typedef float v8f __attribute__((ext_vector_type(8)));
typedef int   gvec4i __attribute__((vector_size(16)));  // matches builtin sig

#define BB 4
#define SS 2048
#define DD 512
#define HH 8
#define DHH 64
#define STILE 16
#define SCALE 0.125f /* DH^-0.5 = 1/8 */

#define XPAD (DD + 4)   /* 516: (4n+j)%64 distinct -> conflict-free, 16B-mult */
#define KPAD (DHH + 4)  /* 68 */
#define VPAD 20         /* 16-float rows padded to 20 */
#define PPAD 20

#define AS1 __attribute__((address_space(1)))
#define AS3 __attribute__((address_space(3)))

// 16B global -> LDS async copy (GLOBAL_LOAD_ASYNC_TO_LDS_B128, ASYNCcnt).
static __device__ __forceinline__ void async_copy16(const float* g, float* l) {
  __builtin_amdgcn_global_load_async_to_lds_b128((AS1 gvec4i*)(void*)g,
                                                 (AS3 gvec4i*)(void*)l, 0, 0);
}

template <int N>
static __device__ __forceinline__ void async_wait() {
#if __has_builtin(__builtin_amdgcn_s_wait_asynccnt)
  __builtin_amdgcn_s_wait_asynccnt(N);
#else
  asm volatile("s_wait_asynccnt %0" ::"i"(N) : "memory");
#endif
}

static __device__ __forceinline__ v8f wmma_f32x4(v2f a, v2f b, v8f c) {
  // D = A(16x4) * B(4x16) + C(16x16), fp32.
  return __builtin_amdgcn_wmma_f32_16x16x4_f32(false, a, false, b, (short)0, c,
                                               false, false);
}

static __device__ __forceinline__ v2f load2(const float* p) {
  return *(const v2f*)p;
}

// ---------------------------------------------------------------------------
// Kernel 0: transpose W [H][D][DH] -> Wt [H][DH][D] (LDS-tiled, 32x32).
// grid = (32, H, 3), block = 256 (32x8 logical).
// Makes the WMMA B-fragment pair (W[k][n], W[k+1][n]) a contiguous b64 load.
// ---------------------------------------------------------------------------
__global__ void __launch_bounds__(256)
mha_transpose_w(const float* __restrict__ Wq, const float* __restrict__ Wk,
                const float* __restrict__ Wv, float* __restrict__ Wqt,
                float* __restrict__ Wkt, float* __restrict__ Wvt) {
  __shared__ float tileT[32][33];
  const int which = blockIdx.z;
  const float* src = (which == 0) ? Wq : (which == 1) ? Wk : Wv;
  float* dst = (which == 0) ? Wqt : (which == 1) ? Wkt : Wvt;
  const int h  = blockIdx.y;
  const int td = (blockIdx.x / (DHH / 32)) * 32;  // D-tile origin
  const int tn = (blockIdx.x % (DHH / 32)) * 32;  // DH-tile origin
  const int tx = threadIdx.x & 31;
  const int ty = threadIdx.x >> 5;

  const float* s = src + ((size_t)h * DD + td) * DHH + tn;
  float* d = dst + ((size_t)h * DHH + tn) * DD + td;
#pragma unroll
  for (int r = 0; r < 32; r += 8)
    tileT[tx][ty + r] = s[(size_t)(ty + r) * DHH + tx];  // coalesced read
  __syncthreads();
#pragma unroll
  for (int r = 0; r < 32; r += 8)
    d[(size_t)(ty + r) * DD + tx] = tileT[ty + r][tx];   // coalesced write
}

// ---------------------------------------------------------------------------
// Kernel 1: Q/K/V projection.
// grid = B*S/16 blocks, block = 256 (8 waves). Wave w == head w.
// The 16(S) x 512(D) x-tile is async-staged into padded LDS once and shared
// by all 8 heads. Q/K stored [B][H][S][DH]; V stored transposed [B][H][DH][S]
// so the attention P*V B-fragments become contiguous b64 loads.
// Occupancy is LDS-limited, so launch_bounds(...,1) frees the register file
// for memory-level parallelism.
// ---------------------------------------------------------------------------
__global__ void __launch_bounds__(256, 1)
mha_qkv_proj(const float* __restrict__ x,
             const float* __restrict__ Wqt,
             const float* __restrict__ Wkt,
             const float* __restrict__ Wvt,
             float* __restrict__ qo,
             float* __restrict__ ko,
             float* __restrict__ vto) {
  __shared__ float Xlds[STILE * XPAD];  // ~33 KB, padded rows

  const int tid   = threadIdx.x;
  const int lane  = tid & 31;
  const int h     = tid >> 5;          // wave index == head
  const int mrow  = lane & 15;         // A-matrix row / B,C column index
  const int khalf = (lane >> 4) * 2;   // K sub-offset per lane half
  const int ncol  = lane & 15;

  const int b  = blockIdx.x / (SS / STILE);
  const int s0 = (blockIdx.x % (SS / STILE)) * STILE;

  // Cooperative async load of the 16x512 x tile into padded LDS rows.
  const float* gx = x + ((size_t)b * SS + s0) * DD;
#pragma unroll
  for (int p = 0; p < 8; ++p) {
    const int c  = p * 256 + tid;  // 16B chunk id, 128 chunks per row
    const int r  = c >> 7;
    const int cc = c & 127;
    async_copy16(gx + (size_t)r * DD + cc * 4, &Xlds[r * XPAD + cc * 4]);
  }
  async_wait<0>();
  __syncthreads();

  const float* wq = Wqt + (size_t)h * DHH * DD;  // [DH][D]
  const float* wk = Wkt + (size_t)h * DHH * DD;
  const float* wv = Wvt + (size_t)h * DHH * DD;

  v8f accq[4] = {v8f{}, v8f{}, v8f{}, v8f{}};
  v8f acck[4] = {v8f{}, v8f{}, v8f{}, v8f{}};
  v8f accv[4] = {v8f{}, v8f{}, v8f{}, v8f{}};

  const float* xrow = &Xlds[mrow * XPAD];
  for (int k0 = 0; k0 < DD; k0 += 4) {
    // ---- load phase: A fragment + all 12 B fragments (clause together) ----
    const v2f a = load2(&xrow[k0 + khalf]);  // ds_load_b64, conflict-free
    v2f bq[4], bk[4], bv[4];
#pragma unroll
    for (int t = 0; t < 4; ++t) {
      const size_t ncl = (size_t)(t * 16 + ncol) * DD + k0 + khalf;
      bq[t] = load2(&wq[ncl]);
      bk[t] = load2(&wk[ncl]);
      bv[t] = load2(&wv[ncl]);
    }
    // ---- compute phase: 12 WMMAs drain with staggered loadcnt waits ----
#pragma unroll
    for (int t = 0; t < 4; ++t) {
      accq[t] = wmma_f32x4(a, bq[t], accq[t]);
      acck[t] = wmma_f32x4(a, bk[t], acck[t]);
      accv[t] = wmma_f32x4(a, bv[t], accv[t]);
    }
  }

  // Store. C layout: lane holds rows (vi + (lane/16)*8), col = lane&15.
  const size_t base  = (((size_t)b * HH + h) * SS + s0) * DHH;
  const size_t vbase = ((size_t)b * HH + h) * DHH * SS;  // transposed V
#pragma unroll
  for (int t = 0; t < 4; ++t) {
#pragma unroll
    for (int vi = 0; vi < 8; ++vi) {
      const int m = vi + (lane >> 4) * 8;
      const int n = t * 16 + ncol;
      qo[base + (size_t)m * DHH + n] = accq[t][vi];
      ko[base + (size_t)m * DHH + n] = acck[t][vi];
      vto[vbase + (size_t)n * SS + s0 + m] = accv[t][vi];  // merges over vi
    }
  }
}

// ---------------------------------------------------------------------------
// Kernel 2: flash attention.
// grid = B*H*(S/128), block = 256 (8 waves). Each wave owns one 16-query
// tile; waves share double-buffered K (16x64) and Vt (64x16) tiles that are
// async-prefetched into padded LDS (tile i+1 in flight while computing i).
// Output: [B][S][H*DH] fp32 (concat heads).
// ---------------------------------------------------------------------------
__global__ void __launch_bounds__(256, 1)
mha_flash_attn(const float* __restrict__ q,
               const float* __restrict__ k,
               const float* __restrict__ vt,
               float* __restrict__ out) {
  __shared__ float Kbuf[2][STILE * KPAD];  // 2 x 16x68
  __shared__ float Vbuf[2][DHH * VPAD];    // 2 x 64x20 (transposed V tile)
  __shared__ float Pall[8][16 * PPAD];     // per-wave padded P tile

  const int tid   = threadIdx.x;
  const int lane  = tid & 31;
  const int w     = tid >> 5;
  const int mrow  = lane & 15;
  const int khalf = (lane >> 4) * 2;
  const int ncol  = lane & 15;
  float* Pld = &Pall[w][0];

  int idx = blockIdx.x;
  const int bt = idx % (SS / 128); idx /= (SS / 128);
  const int h  = idx % HH;
  const int b  = idx / HH;
  const int s0q = (bt * 8 + w) * STILE;  // this wave's query tile

  const float* qbase  = q  + (((size_t)b * HH + h) * SS) * DHH;
  const float* kbase  = k  + (((size_t)b * HH + h) * SS) * DHH;
  const float* vtbase = vt + ((size_t)b * HH + h) * DHH * SS;

  // Per-thread staging coordinates.
  const int kr = tid >> 4, kc = tid & 15;  // K: 16 rows x 16 chunks
  const int vr = tid >> 2, vc = tid & 3;   // Vt: 64 rows x 4 chunks

  // Q tile A-fragments: 16 K-steps of 4 over DH=64 (kept in registers).
  v2f qa[16];
  {
    const float* qrow = qbase + (size_t)(s0q + mrow) * DHH;
#pragma unroll
    for (int kk = 0; kk < 16; ++kk) qa[kk] = load2(&qrow[kk * 4 + khalf]);
  }

  v8f acco[4] = {v8f{}, v8f{}, v8f{}, v8f{}};
  float rowmax[8], rowsum[8];
#pragma unroll
  for (int vi = 0; vi < 8; ++vi) { rowmax[vi] = -1.0e30f; rowsum[vi] = 0.0f; }

  // Prefetch tile 0.
  async_copy16(kbase + (size_t)kr * DHH + kc * 4, &Kbuf[0][kr * KPAD + kc * 4]);
  async_copy16(vtbase + (size_t)vr * SS + vc * 4, &Vbuf[0][vr * VPAD + vc * 4]);

  for (int it = 0; it < SS / STILE; ++it) {
    const int cur = it & 1;

    // Prefetch tile it+1 into the other buffer, then wait for tile it.
    if (it + 1 < SS / STILE) {
      const float* kn = kbase + (size_t)(it + 1) * STILE * DHH;
      const float* vn = vtbase + (it + 1) * STILE;
      async_copy16(kn + (size_t)kr * DHH + kc * 4,
                   &Kbuf[cur ^ 1][kr * KPAD + kc * 4]);
      async_copy16(vn + (size_t)vr * SS + vc * 4,
                   &Vbuf[cur ^ 1][vr * VPAD + vc * 4]);
      async_wait<2>();  // per-wave, in-order: tile it's 2 asyncs are done
    } else {
      async_wait<0>();
    }
    __syncthreads();

    const float* Kl = &Kbuf[cur][0];
    const float* Vl = &Vbuf[cur][0];

    // ---- score tile: S = Q * K^T (16x16) over DH, fp32 WMMA ----
    v2f bk[16];
#pragma unroll
    for (int kk = 0; kk < 16; ++kk)
      bk[kk] = load2(&Kl[mrow * KPAD + kk * 4 + khalf]);
    v8f sc = v8f{};
#pragma unroll
    for (int kk = 0; kk < 16; ++kk) sc = wmma_f32x4(qa[kk], bk[kk], sc);

    // ---- online softmax (rows live across a 16-lane half) ----
    float alpha[8];
#pragma unroll
    for (int vi = 0; vi < 8; ++vi) {
      float s = sc[vi] * SCALE;
      float tmax = s;
      tmax = fmaxf(tmax, __shfl_xor(tmax, 1));
      tmax = fmaxf(tmax, __shfl_xor(tmax, 2));
      tmax = fmaxf(tmax, __shfl_xor(tmax, 4));
      tmax = fmaxf(tmax, __shfl_xor(tmax, 8));
      const float nm = fmaxf(rowmax[vi], tmax);
      alpha[vi] = __expf(rowmax[vi] - nm);
      const float p = __expf(s - nm);
      float psum = p;
      psum += __shfl_xor(psum, 1);
      psum += __shfl_xor(psum, 2);
      psum += __shfl_xor(psum, 4);
      psum += __shfl_xor(psum, 8);
      rowsum[vi] = rowsum[vi] * alpha[vi] + psum;
      rowmax[vi] = nm;
      // stash P (C-layout position) in this wave's private padded LDS slice
      Pld[(vi + (lane >> 4) * 8) * PPAD + ncol] = p;
    }
#pragma unroll
    for (int t = 0; t < 4; ++t)
#pragma unroll
      for (int vi = 0; vi < 8; ++vi) acco[t][vi] *= alpha[vi];

    // ---- O += P * V : A = P (4 K-steps), B = Vt tile (b64 loads) ----
    v2f pa[4], bv[16];
#pragma unroll
    for (int ks = 0; ks < 4; ++ks)
      pa[ks] = load2(&Pld[mrow * PPAD + ks * 4 + khalf]);
#pragma unroll
    for (int ks = 0; ks < 4; ++ks)
#pragma unroll
      for (int t = 0; t < 4; ++t)
        bv[ks * 4 + t] = load2(&Vl[(t * 16 + ncol) * VPAD + ks * 4 + khalf]);
#pragma unroll
    for (int ks = 0; ks < 4; ++ks)
#pragma unroll
      for (int t = 0; t < 4; ++t)
        acco[t] = wmma_f32x4(pa[ks], bv[ks * 4 + t], acco[t]);

    __syncthreads();  // all waves done with buf[cur] before it is refilled
  }

  // ---- epilogue: divide by row sums, write concat-head output ----
#pragma unroll
  for (int t = 0; t < 4; ++t) {
#pragma unroll
    for (int vi = 0; vi < 8; ++vi) {
      const int m = vi + (lane >> 4) * 8;
      const int n = t * 16 + ncol;
      out[((size_t)b * SS + (s0q + m)) * (HH * DHH) + h * DHH + n] =
          acco[t][vi] / rowsum[vi];
    }
  }
}

extern "C" void kernel_launch(void* const* d_in, const int* in_sizes, int n_in,
                              void* d_out, int out_size, void* d_ws, size_t ws_size,
                              hipStream_t stream) {
  const float* x  = (const float*)d_in[0];
  const float* Wq = (const float*)d_in[1];
  const float* Wk = (const float*)d_in[2];
  const float* Wv = (const float*)d_in[3];
  float* out = (float*)d_out;

  const size_t per  = (size_t)BB * HH * SS * DHH;  // 4,194,304 floats
  const size_t wper = (size_t)HH * DD * DHH;       //   262,144 floats
  float* qws  = (float*)d_ws;
  float* kws  = qws + per;
  float* vtws = kws + per;
  float* wqt  = vtws + per;
  float* wkt  = wqt + wper;
  float* wvt  = wkt + wper;

  dim3 gT(32, HH, 3);  // (D/32)*(DH/32)=32 tiles per head per matrix
  mha_transpose_w<<<gT, 256, 0, stream>>>(Wq, Wk, Wv, wqt, wkt, wvt);

  dim3 g1(BB * (SS / STILE));          // 512 blocks x 256 threads
  mha_qkv_proj<<<g1, 256, 0, stream>>>(x, wqt, wkt, wvt, qws, kws, vtws);

  dim3 g2(BB * HH * (SS / 128));       // 512 blocks x 256 threads
  mha_flash_attn<<<g2, 256, 0, stream>>>(qws, kws, vtws, out);
}